// LinkPredictor_87900800680117
// MI455X (gfx1250) — compile-verified
//
#include <hip/hip_runtime.h>

// ---------------------------------------------------------------------------
// GraphSAGE 2-layer forward for MI455X (gfx1250), fp32 WMMA 16x16x4 path.
//
// N=100000 nodes, E=200000 edges, IN_DIM=384, HIDDEN=128. Output [N,128] f32.
//
// Pipeline (all on `stream`, graph-capture safe):
//   0. zero deg + agg1 region in ws; pack 4 weight matrices into WMMA
//      B-fragment layout (device-side, tiny).
//   1. degree[dst] += 1 per edge (atomic f32)
//   2. agg1[dst,:384] += x[src,:384]  (float4 atomics)
//   3. GEMM pass A: h1  = (agg1 * 1/max(deg,1)) @ W1_l + b1
//   4. GEMM pass B: h1  = relu(h1 + x @ W1_r)
//   5. zero agg2 (reuse agg1 buffer), agg2[dst,:128] += h1[src,:128]
//   6. GEMM pass C: out = (agg2 * 1/max(deg,1)) @ W2_l + b2
//   7. GEMM pass D: out = out + h1 @ W2_r
//
// GEMM inner loop is software-pipelined: A fragment for k-step t+1 is
// prefetched before the 8-WMMA batch of k-step t; all 8 B fragments are
// pulled from LDS into registers before the WMMA batch so ds waits amortize.
// ---------------------------------------------------------------------------

typedef __attribute__((ext_vector_type(2))) float v2f;
typedef __attribute__((ext_vector_type(8))) float v8f;

#define TILES_PER_BLOCK 8   // 8 waves per block, one 16-row tile per wave
#define CHUNK_KT 32         // 32 k-steps (K=128 values) of packed B per LDS stage

// ---------------------------------------------------------------------------
__global__ __launch_bounds__(256) void zero_f32_kernel(float* __restrict__ p,
                                                       long n) {
  long i = (long)blockIdx.x * blockDim.x + threadIdx.x;
  long stride = (long)gridDim.x * blockDim.x;
  for (; i < n; i += stride) p[i] = 0.0f;
}

// Pack row-major W[K,128] into per-fragment lane layout for
// V_WMMA_F32_16X16X4_F32 B operand (4x16 tile, 2 VGPRs/lane):
//   P[(kt*8+nt)*64 + L*2 + v] = W[kt*4 + (L>>4)*2 + v][nt*16 + (L&15)]
__global__ __launch_bounds__(256) void pack_weights_kernel(
    const float* __restrict__ W, float* __restrict__ P, int K) {
  int total = K * 128;
  for (int tid = blockIdx.x * blockDim.x + threadIdx.x; tid < total;
       tid += gridDim.x * blockDim.x) {
    int frag = tid >> 6;   // which 4x16 fragment
    int r = tid & 63;
    int L = r >> 1;        // lane
    int v = r & 1;         // vgpr within fragment
    int kt = frag >> 3;
    int nt = frag & 7;
    int k = kt * 4 + ((L >> 4) << 1) + v;
    int n = nt * 16 + (L & 15);
    P[tid] = W[k * 128 + n];
  }
}

__global__ __launch_bounds__(256) void degree_kernel(
    const int* __restrict__ dst, float* __restrict__ deg, int E) {
  int e = blockIdx.x * blockDim.x + threadIdx.x;
  if (e < E) atomicAdd(&deg[dst[e]], 1.0f);
}

// agg[dst[e], :] += feat[src[e], :]   (F multiple of 4)
__global__ __launch_bounds__(256) void scatter_add_kernel(
    const float* __restrict__ feat, const int* __restrict__ src,
    const int* __restrict__ dst, float* __restrict__ agg, int E, int F) {
  int chunks = F >> 2;
  long total = (long)E * chunks;
  long tid = (long)blockIdx.x * blockDim.x + threadIdx.x;
  long stride = (long)gridDim.x * blockDim.x;
  for (; tid < total; tid += stride) {
    int e = (int)(tid / chunks);
    int c = (int)(tid % chunks);
    int s = src[e];
    int d = dst[e];
    const float4 val = *(const float4*)(feat + (size_t)s * F + c * 4);
    float* o = agg + (size_t)d * F + c * 4;
    atomicAdd(o + 0, val.x);
    atomicAdd(o + 1, val.y);
    atomicAdd(o + 2, val.z);
    atomicAdd(o + 3, val.w);
  }
}

// ---------------------------------------------------------------------------
// One GEMM pass: out[tile rows, 128] = f(A_scaled @ Bpacked + bias + prev)
//   A: [nTiles*16, K] f32, K in {128,384} (multiple of 128)
//   Bpacked: (K/4)*8 fragments of 64 floats (see pack_weights_kernel)
//   deg != nullptr  -> scale A row by 1/max(deg,1)  (fused mean)
//   prev != nullptr -> add prev[m*128+n] (may alias out; same-thread RMW)
//   relu != 0       -> ReLU epilogue
// Each wave computes a 16x128 tile; 8x v_wmma_f32_16x16x4_f32 per k-step.
// ---------------------------------------------------------------------------
__global__ __launch_bounds__(256) void gemm_pass_kernel(
    const float* __restrict__ A, int K, const float* __restrict__ Bp,
    const float* __restrict__ bias, const float* __restrict__ deg,
    const float* __restrict__ prev, float* __restrict__ out, int relu,
    int nTiles) {
  __shared__ float lds[CHUNK_KT * 8 * 64];  // 64 KB staged B fragments

  const int lane = threadIdx.x & 31;
  const int wave = threadIdx.x >> 5;
  int tile = blockIdx.x * TILES_PER_BLOCK + wave;
  const bool active = tile < nTiles;
  if (!active) tile = nTiles - 1;  // keep wave converged through WMMA/barriers

  const int half = lane >> 4;  // lane half selects K sub-pair / M offset
  const int n0 = lane & 15;    // column within 16-wide tile / A row within tile
  const int arow = tile * 16 + n0;
  const float* __restrict__ arowp = A + (size_t)arow * K + half * 2;

  float scale = 1.0f;
  if (deg) scale = 1.0f / fmaxf(deg[arow], 1.0f);

  v8f acc[8];
#pragma unroll
  for (int nt = 0; nt < 8; ++nt) {
    float b = bias ? bias[nt * 16 + n0] : 0.0f;
    acc[nt] = (v8f){b, b, b, b, b, b, b, b};
  }

  const int nkt = K >> 2;  // number of k-steps of 4
  // software pipeline: prefetch A fragment for k-step 0
  v2f a_pre = *(const v2f*)(arowp);

  for (int kt0 = 0; kt0 < nkt; kt0 += CHUNK_KT) {
    __syncthreads();
    // stage CHUNK_KT k-steps of packed B fragments (float4-vectorized copy)
    {
      const float4* __restrict__ gsrc =
          (const float4*)(Bp + (size_t)kt0 * 8 * 64);
      float4* __restrict__ ldst = (float4*)lds;
      for (int i = threadIdx.x; i < CHUNK_KT * 8 * 16; i += 256)
        ldst[i] = gsrc[i];
    }
    __syncthreads();

    for (int kt = 0; kt < CHUNK_KT; ++kt) {
      const int ktg = kt0 + kt;
      // consume prefetched A fragment; apply fused mean scale
      v2f a = a_pre * scale;
      // issue prefetch of the next A fragment (clamped at the end)
      const int ktn = (ktg + 1 < nkt) ? (ktg + 1) : ktg;
      a_pre = *(const v2f*)(arowp + ktn * 4);

      // pull all 8 B fragments into registers first (ds loads overlap)
      const float* bbase = lds + (kt * 8) * 64 + lane * 2;
      v2f bf[8];
#pragma unroll
      for (int nt = 0; nt < 8; ++nt) bf[nt] = *(const v2f*)(bbase + nt * 64);

      // back-to-back WMMA batch
#pragma unroll
      for (int nt = 0; nt < 8; ++nt) {
        acc[nt] = __builtin_amdgcn_wmma_f32_16x16x4_f32(
            false, a, false, bf[nt], (short)0, acc[nt], false, false);
      }
    }
  }

  if (!active) return;

  // C/D layout: vgpr r -> M = r + 8*half, lane n0 -> N = nt*16 + n0
  const int m0 = tile * 16;
#pragma unroll
  for (int nt = 0; nt < 8; ++nt) {
#pragma unroll
    for (int r = 0; r < 8; ++r) {
      int m = m0 + r + 8 * half;
      size_t idx = (size_t)m * 128 + nt * 16 + n0;
      float v = acc[nt][r];
      if (prev) v += prev[idx];
      if (relu) v = fmaxf(v, 0.0f);
      out[idx] = v;
    }
  }
}

// ---------------------------------------------------------------------------
extern "C" void kernel_launch(void* const* d_in, const int* in_sizes, int n_in,
                              void* d_out, int out_size, void* d_ws,
                              size_t ws_size, hipStream_t stream) {
  const int IN_DIM = 384, HID = 128;
  const int N = in_sizes[0] / IN_DIM;   // 100000
  const int E = in_sizes[1] / 2;        // 200000

  const float* x    = (const float*)d_in[0];
  const int*   ei   = (const int*)d_in[1];   // [2,E] int32
  const float* W1_l = (const float*)d_in[2];
  const float* b1   = (const float*)d_in[3];
  const float* W1_r = (const float*)d_in[4];
  const float* W2_l = (const float*)d_in[5];
  const float* b2   = (const float*)d_in[6];
  const float* W2_r = (const float*)d_in[7];
  const int* src = ei;
  const int* dst = ei + E;
  float* out = (float*)d_out;

  // workspace layout (floats)
  float* ws = (float*)d_ws;
  const long deg_off  = 0;
  const long deg_pad  = ((long)N + 127) & ~127L;          // 100096
  const long agg_off  = deg_off + deg_pad;                // N*384 region
  const long h1_off   = agg_off + (long)N * IN_DIM;
  const long pw1l_off = h1_off + (long)N * HID;
  const long pw1r_off = pw1l_off + (long)IN_DIM * HID;
  const long pw2l_off = pw1r_off + (long)IN_DIM * HID;
  const long pw2r_off = pw2l_off + (long)HID * HID;

  float* deg  = ws + deg_off;
  float* agg  = ws + agg_off;   // reused as agg2 ([N,128]) for layer 2
  float* h1   = ws + h1_off;
  float* pw1l = ws + pw1l_off;
  float* pw1r = ws + pw1r_off;
  float* pw2l = ws + pw2l_off;
  float* pw2r = ws + pw2r_off;

  const int nTiles = N / 16;  // 6250 (N divisible by 16)
  const int gemmBlocks = (nTiles + TILES_PER_BLOCK - 1) / TILES_PER_BLOCK;

  // 0) zero deg + agg1 (contiguous), pack weights
  {
    long nz = deg_pad + (long)N * IN_DIM;
    int blocks = (int)((nz + 255) / 256);
    zero_f32_kernel<<<blocks, 256, 0, stream>>>(ws, nz);
  }
  pack_weights_kernel<<<(IN_DIM * 128 + 255) / 256, 256, 0, stream>>>(W1_l, pw1l, IN_DIM);
  pack_weights_kernel<<<(IN_DIM * 128 + 255) / 256, 256, 0, stream>>>(W1_r, pw1r, IN_DIM);
  pack_weights_kernel<<<(HID * 128 + 255) / 256, 256, 0, stream>>>(W2_l, pw2l, HID);
  pack_weights_kernel<<<(HID * 128 + 255) / 256, 256, 0, stream>>>(W2_r, pw2r, HID);

  // 1) degrees
  degree_kernel<<<(E + 255) / 256, 256, 0, stream>>>(dst, deg, E);

  // 2) scatter x -> agg1
  {
    long total = (long)E * (IN_DIM / 4);
    int blocks = (int)((total + 255) / 256);
    scatter_add_kernel<<<blocks, 256, 0, stream>>>(x, src, dst, agg, E, IN_DIM);
  }

  // 3) h1 = (agg1/deg) @ W1_l + b1
  gemm_pass_kernel<<<gemmBlocks, 256, 0, stream>>>(agg, IN_DIM, pw1l, b1, deg,
                                                   nullptr, h1, 0, nTiles);
  // 4) h1 = relu(h1 + x @ W1_r)
  gemm_pass_kernel<<<gemmBlocks, 256, 0, stream>>>(x, IN_DIM, pw1r, nullptr,
                                                   nullptr, h1, h1, 1, nTiles);

  // 5) zero agg2, scatter h1 -> agg2
  {
    long nz = (long)N * HID;
    int blocks = (int)((nz + 255) / 256);
    zero_f32_kernel<<<blocks, 256, 0, stream>>>(agg, nz);
    long total = (long)E * (HID / 4);
    int sblocks = (int)((total + 255) / 256);
    scatter_add_kernel<<<sblocks, 256, 0, stream>>>(h1, src, dst, agg, E, HID);
  }

  // 6) out = (agg2/deg) @ W2_l + b2
  gemm_pass_kernel<<<gemmBlocks, 256, 0, stream>>>(agg, HID, pw2l, b2, deg,
                                                   nullptr, out, 0, nTiles);
  // 7) out = out + h1 @ W2_r
  gemm_pass_kernel<<<gemmBlocks, 256, 0, stream>>>(h1, HID, pw2r, nullptr,
                                                   nullptr, out, out, 0, nTiles);
}